// ODENetDerivative_7447473291343
// MI455X (gfx1250) — compile-verified
//
#include <hip/hip_runtime.h>

typedef __attribute__((ext_vector_type(2))) float v2f;
typedef __attribute__((ext_vector_type(8))) float v8f;

// Broadcast value held by lane (srcLane in 0..15) across the wave32.
__device__ __forceinline__ float lane_bcast(float v, int srcLane) {
    return __shfl(v, srcLane, 32);
}

// One wave = 16 batch elements.
// Each 4-wide layer is D = W_pad(16x4) x h^T(4x16) on the matrix pipe:
//   A (16x4 f32): lane m holds A[m][k0], A[m][k0+1]; k0 = (lane<16)?0:2
//   B (4x16 f32): vgpr0 = {K=0 | K=2}, vgpr1 = {K=1 | K=3}, N = lane&15
//   D (16x16 f32): feature row r -> vgpr r, lanes 0..15 (rows 4..15 are zero pad)
__global__ void __launch_bounds__(256)
odenet_wmma_kernel(const float* __restrict__ t, const float* __restrict__ y,
                   const float* __restrict__ inp, const float* __restrict__ dt,
                   const float* __restrict__ W1, const float* __restrict__ W2,
                   const float* __restrict__ W3, const float* __restrict__ W4,
                   const float* __restrict__ wscale, float* __restrict__ out,
                   int Btot, int frames) {
    const int lane = threadIdx.x & 31;
    const int wave = (blockIdx.x * blockDim.x + threadIdx.x) >> 5;
    const int n    = lane & 15;        // batch column within wave / feature row select
    const int hi   = lane >> 4;        // 0: K=0,1 half   1: K=2,3 half
    const int base = wave * 16;
    if (base >= Btot) return;          // wave-uniform: EXEC stays all-1s for live waves

    // idx = int(t/dt) % frames, computed in f32 like the reference
    const float tv  = t[0];
    const float dtv = dt[0];
    const int   idx = ((int)(tv / dtv)) % frames;

    const int col = base + n;
    const float yv = y[col];
    const float xv = inp[(long long)idx * (long long)Btot + col];

    const int  mm  = n & 3;            // always-in-bounds weight row index
    const bool mok = (n < 4);          // valid weight row (rows 4..15 are zero pad)
    const int  k0  = hi ? 2 : 0;

    const v8f zero = {0.f, 0.f, 0.f, 0.f, 0.f, 0.f, 0.f, 0.f};

    // ---- Layer 1: h1 = tanh([y, x] @ W1^T), W1 is [4,2] (K=2,3 zero-padded) ----
    v2f A, Bm;
    {
        const float w1a = W1[mm * 2 + 0];
        const float w1b = W1[mm * 2 + 1];
        A[0]  = (mok && !hi) ? w1a : 0.0f;   // K=2,3 of W1 don't exist -> 0
        A[1]  = (mok && !hi) ? w1b : 0.0f;
        Bm[0] = hi ? 0.0f : yv;              // K=0 row = y   | K=2 row = 0
        Bm[1] = hi ? 0.0f : xv;              // K=1 row = x   | K=3 row = 0
    }
    v8f d = __builtin_amdgcn_wmma_f32_16x16x4_f32(false, A, false, Bm,
                                                  (short)0, zero, false, false);
    // repack D rows {0..3} into B layout for next layer, apply tanh
    float s2 = lane_bcast(d[2], n);
    float s3 = lane_bcast(d[3], n);
    float b0 = tanhf(hi ? s2 : d[0]);
    float b1 = tanhf(hi ? s3 : d[1]);

    // ---- Layer 2: h2 = tanh(h1 @ W2^T), W2 is [4,4] ----
    {
        const float w2a = W2[mm * 4 + k0];
        const float w2b = W2[mm * 4 + k0 + 1];
        A[0]  = mok ? w2a : 0.0f;
        A[1]  = mok ? w2b : 0.0f;
        Bm[0] = b0;
        Bm[1] = b1;
    }
    d  = __builtin_amdgcn_wmma_f32_16x16x4_f32(false, A, false, Bm,
                                               (short)0, zero, false, false);
    s2 = lane_bcast(d[2], n);
    s3 = lane_bcast(d[3], n);
    b0 = tanhf(hi ? s2 : d[0]);
    b1 = tanhf(hi ? s3 : d[1]);

    // ---- Layer 3: h3 = tanh(h2 @ W3^T), W3 is [4,4] ----
    {
        const float w3a = W3[mm * 4 + k0];
        const float w3b = W3[mm * 4 + k0 + 1];
        A[0]  = mok ? w3a : 0.0f;
        A[1]  = mok ? w3b : 0.0f;
        Bm[0] = b0;
        Bm[1] = b1;
    }
    d  = __builtin_amdgcn_wmma_f32_16x16x4_f32(false, A, false, Bm,
                                               (short)0, zero, false, false);
    s2 = lane_bcast(d[2], n);
    s3 = lane_bcast(d[3], n);
    b0 = tanhf(hi ? s2 : d[0]);
    b1 = tanhf(hi ? s3 : d[1]);

    // ---- Layer 4 + scale: out = (h3 @ W4^T) * w_scale, W4 is [1,4] ----
    // lane n holds partial over K={0,1}, lane n+16 holds partial over K={2,3}
    const float p  = W4[k0] * b0 + W4[k0 + 1] * b1;
    const float ph = __shfl(p, n + 16, 32);   // hi-half partial into lanes 0..15
    if (!hi) out[col] = (p + ph) * wscale[0];
}

extern "C" void kernel_launch(void* const* d_in, const int* in_sizes, int n_in,
                              void* d_out, int out_size, void* d_ws, size_t ws_size,
                              hipStream_t stream) {
    // setup_inputs() order: t, y, inp, dt, W1, W2, W3, W4, w_scale
    const float* t   = (const float*)d_in[0];
    const float* y   = (const float*)d_in[1];
    const float* inp = (const float*)d_in[2];
    const float* dt  = (const float*)d_in[3];
    const float* W1  = (const float*)d_in[4];
    const float* W2  = (const float*)d_in[5];
    const float* W3  = (const float*)d_in[6];
    const float* W4  = (const float*)d_in[7];
    const float* wsc = (const float*)d_in[8];
    float* out = (float*)d_out;

    const int B      = in_sizes[1];          // 1024
    const int frames = in_sizes[2] / B;      // 44100

    const int waves   = (B + 15) / 16;       // 64 waves, 16 batch elems each
    const int threads = waves * 32;          // wave32
    const int block   = 256;                 // 8 waves per workgroup
    const int grid    = (threads + block - 1) / block;

    odenet_wmma_kernel<<<grid, block, 0, stream>>>(t, y, inp, dt, W1, W2, W3, W4,
                                                   wsc, out, B, frames);
}